// POGCN_64802466562600
// MI455X (gfx1250) — compile-verified
//
#include <hip/hip_runtime.h>
#include <hip/hip_bf16.h>
#include <stdint.h>

#define NUM_USERS 150000
#define NUM_ITEMS 80000
#define N_NODES   230000
#define NNZ_TOTAL 5000000
#define DIM       64
#define N_LAYERS  3

#define TILE       2048
#define BLK        256
#define SCAN_CHUNK 1024
#define NBLK_SCAN1 ((N_NODES + SCAN_CHUNK - 1) / SCAN_CHUNK)   // 225
static_assert(NBLK_SCAN1 <= 256, "scan2 assumes one block");

// ---------------- CDNA5-specific primitives (gfx1250) ----------------

// Async global -> LDS copy, 16B per lane. Tracked by ASYNCcnt. (verified to assemble)
__device__ __forceinline__ void async_b128(uint32_t lds_off, const void* gaddr) {
  asm volatile("global_load_async_to_lds_b128 %0, %1, off"
               :: "v"(lds_off), "v"(gaddr) : "memory");
}
__device__ __forceinline__ void wait_async0() {
  asm volatile("s_wait_asynccnt 0" ::: "memory");
}
// Low 32 bits of a generic pointer to __shared__ == LDS byte address (ISA 10.2).
__device__ __forceinline__ uint32_t lds_addr_of(const void* p) {
  return (uint32_t)(uintptr_t)p;
}

// ---------------- embedding init ----------------

__global__ __launch_bounds__(256) void pogcn_init(const float4* __restrict__ ue,
                                                  const float4* __restrict__ ie,
                                                  float4* __restrict__ cur,
                                                  float4* __restrict__ acc,
                                                  int n4u, int n4tot) {
  int i = blockIdx.x * 256 + threadIdx.x;
  if (i < n4tot) {
    float4 v = (i < n4u) ? ue[i] : ie[i - n4u];
    cur[i] = v;
    acc[i] = v;
  }
}

__global__ __launch_bounds__(256) void pogcn_zero_i4(int4* __restrict__ p, int n4) {
  int i = blockIdx.x * 256 + threadIdx.x;
  if (i < n4) p[i] = make_int4(0, 0, 0, 0);
}

// ---------------- CSR build ----------------

__global__ __launch_bounds__(256) void pogcn_count(const int* __restrict__ rows,
                                                   int* __restrict__ cnt, int nnz) {
  int i = blockIdx.x * 256 + threadIdx.x;
  if (i < nnz) atomicAdd(&cnt[rows[i]], 1);
}

// Phase 1: per-1024-chunk exclusive scan (256 threads x 4 elems), emit block totals.
__global__ __launch_bounds__(256) void pogcn_scan1(const int* __restrict__ cnt,
                                                   int* __restrict__ out,
                                                   int* __restrict__ bsum, int n) {
  __shared__ int ssum[256];
  const int t = threadIdx.x;
  const int base = blockIdx.x * SCAN_CHUNK + t * 4;
  int e0 = (base + 0 < n) ? cnt[base + 0] : 0;
  int e1 = (base + 1 < n) ? cnt[base + 1] : 0;
  int e2 = (base + 2 < n) ? cnt[base + 2] : 0;
  int e3 = (base + 3 < n) ? cnt[base + 3] : 0;
  int my = e0 + e1 + e2 + e3;
  ssum[t] = my;
  __syncthreads();
  for (int off = 1; off < 256; off <<= 1) {
    int add = (t >= off) ? ssum[t - off] : 0;
    __syncthreads();
    ssum[t] += add;
    __syncthreads();
  }
  int run = ssum[t] - my;                 // exclusive prefix of this thread's group
  if (base + 0 < n) out[base + 0] = run;  run += e0;
  if (base + 1 < n) out[base + 1] = run;  run += e1;
  if (base + 2 < n) out[base + 2] = run;  run += e2;
  if (base + 3 < n) out[base + 3] = run;
  if (t == 255) bsum[blockIdx.x] = ssum[255];
}

// Phase 2: exclusive scan of block totals (single block).
__global__ __launch_bounds__(256) void pogcn_scan2(int* __restrict__ bsum, int nb) {
  __shared__ int ssum[256];
  const int t = threadIdx.x;
  int my = (t < nb) ? bsum[t] : 0;
  ssum[t] = my;
  __syncthreads();
  for (int off = 1; off < 256; off <<= 1) {
    int add = (t >= off) ? ssum[t - off] : 0;
    __syncthreads();
    ssum[t] += add;
    __syncthreads();
  }
  if (t < nb) bsum[t] = ssum[t] - my;
}

// Phase 3: add block offsets, seed fill cursors, close rowptr.
__global__ __launch_bounds__(256) void pogcn_scan3(int* __restrict__ rowptr,
                                                   int* __restrict__ rowcur,
                                                   const int* __restrict__ bsum,
                                                   int n, int nnz) {
  int i = blockIdx.x * 256 + threadIdx.x;
  if (i < n) {
    int v = rowptr[i] + bsum[i >> 10];   // SCAN_CHUNK == 1024
    rowptr[i] = v;
    rowcur[i] = v;
  }
  if (i == 0) rowptr[n] = nnz;
}

// Fill CSR records, streaming the edge arrays through LDS with double-buffered
// async-to-LDS copies (ASYNCcnt).
__global__ __launch_bounds__(BLK) void pogcn_fill(
    const int*   __restrict__ rows,
    const int*   __restrict__ cols,
    const float* __restrict__ vals,
    int*         __restrict__ rowcur,
    int2*        __restrict__ csr,
    int nnz) {
  __shared__ __align__(16) int   s_rows[2][TILE];
  __shared__ __align__(16) int   s_cols[2][TILE];
  __shared__ __align__(16) float s_vals[2][TILE];

  const int tid      = threadIdx.x;
  const int numTiles = (nnz + TILE - 1) / TILE;
  const int stride   = gridDim.x;

  auto is_full = [&](int t) { return (t + 1) * TILE <= nnz; };
  auto stage = [&](int t, int b) {
    const int base = t * TILE;
    for (int k = tid; k < TILE / 4; k += BLK) {
      async_b128(lds_addr_of(&s_rows[b][k * 4]), rows + base + k * 4);
      async_b128(lds_addr_of(&s_cols[b][k * 4]), cols + base + k * 4);
      async_b128(lds_addr_of(&s_vals[b][k * 4]), vals + base + k * 4);
    }
  };

  int buf = 0;
  const int t0 = blockIdx.x;
  if (t0 < numTiles && is_full(t0)) stage(t0, 0);

  for (int t = t0; t < numTiles; t += stride) {
    if (is_full(t)) {
      wait_async0();
      __syncthreads();
      int tn = t + stride;
      if (tn < numTiles && is_full(tn)) stage(tn, buf ^ 1);
#pragma unroll 4
      for (int k = tid; k < TILE; k += BLK) {
        int   r = s_rows[buf][k];
        int   c = s_cols[buf][k];
        float v = s_vals[buf][k];
        int pos = atomicAdd(&rowcur[r], 1);
        csr[pos] = make_int2(c, __float_as_int(v));
      }
      buf ^= 1;
    } else {
      const int base = t * TILE;
      int end = base + TILE; if (end > nnz) end = nnz;
      for (int k = base + tid; k < end; k += BLK) {
        int pos = atomicAdd(&rowcur[rows[k]], 1);
        csr[pos] = make_int2(cols[k], __float_as_int(vals[k]));
      }
    }
  }
}

// ---------------- pull-style SpMM, fused accumulator update ----------------
// One wave per output row; lane l owns dims [2l, 2l+1]. No float atomics.
// Row index is forced wave-uniform (readfirstlane) so rowptr/csr record reads
// take the scalar/SMEM path, keeping the vector pipe free for the 256B gathers.

__global__ __launch_bounds__(256) void pogcn_gather(
    const int*  __restrict__ rowptr,
    const int2* __restrict__ csr,
    const float* __restrict__ x,
    float*       __restrict__ y,
    float*       __restrict__ acc,
    float scale, int storeY) {
  const int lane  = threadIdx.x & 31;
  const int wid   = __builtin_amdgcn_readfirstlane(threadIdx.x >> 5);
  const int gwave = blockIdx.x * 8 + wid;
  const int nwave = gridDim.x * 8;

  for (int r = gwave; r < N_NODES; r += nwave) {
    const int ks = rowptr[r];
    const int ke = rowptr[r + 1];
    float ax = 0.f, ay = 0.f;
    int k = ks;
    for (; k + 3 < ke; k += 4) {           // four gathers in flight
      int2 cv0 = csr[k + 0];
      int2 cv1 = csr[k + 1];
      int2 cv2 = csr[k + 2];
      int2 cv3 = csr[k + 3];
      const float2 x0 = *(const float2*)(x + cv0.x * DIM + lane * 2);
      const float2 x1 = *(const float2*)(x + cv1.x * DIM + lane * 2);
      const float2 x2 = *(const float2*)(x + cv2.x * DIM + lane * 2);
      const float2 x3 = *(const float2*)(x + cv3.x * DIM + lane * 2);
      float v0 = __int_as_float(cv0.y);
      float v1 = __int_as_float(cv1.y);
      float v2 = __int_as_float(cv2.y);
      float v3 = __int_as_float(cv3.y);
      ax = fmaf(v0, x0.x, ax); ay = fmaf(v0, x0.y, ay);
      ax = fmaf(v1, x1.x, ax); ay = fmaf(v1, x1.y, ay);
      ax = fmaf(v2, x2.x, ax); ay = fmaf(v2, x2.y, ay);
      ax = fmaf(v3, x3.x, ax); ay = fmaf(v3, x3.y, ay);
    }
    for (; k < ke; ++k) {
      int2 cv = csr[k];
      const float2 xv = *(const float2*)(x + cv.x * DIM + lane * 2);
      float v = __int_as_float(cv.y);
      ax = fmaf(v, xv.x, ax); ay = fmaf(v, xv.y, ay);
    }
    if (storeY) {
      *(float2*)(y + r * DIM + lane * 2) = make_float2(ax, ay);
    }
    float2* ap = (float2*)(acc + r * DIM + lane * 2);
    float2 av = *ap;
    av.x = (av.x + ax) * scale;
    av.y = (av.y + ay) * scale;
    *ap = av;
  }
}

// ---------------- launcher ----------------

extern "C" void kernel_launch(void* const* d_in, const int* in_sizes, int n_in,
                              void* d_out, int out_size, void* d_ws, size_t ws_size,
                              hipStream_t stream) {
  const float* ue   = (const float*)d_in[0];
  const float* ie   = (const float*)d_in[1];
  const int*   rows = (const int*)d_in[2];
  const int*   cols = (const int*)d_in[3];
  const float* vals = (const float*)d_in[4];

  float* acc = (float*)d_out;                               // N_NODES*DIM floats

  char* ws = (char*)d_ws;
  float* A      = (float*)ws;  ws += (size_t)N_NODES * DIM * 4;   // 58.88 MB
  float* B      = (float*)ws;  ws += (size_t)N_NODES * DIM * 4;   // 58.88 MB
  int2*  csr    = (int2*)ws;   ws += (size_t)NNZ_TOTAL * 8;       // 40.00 MB
  int*   rowptr = (int*)ws;    ws += ((size_t)N_NODES + 16) * 4;  // N+1 (padded)
  int*   rowcur = (int*)ws;    ws += (size_t)N_NODES * 4;         // counts, then cursors
  int*   bsum   = (int*)ws;                                       // scan block sums

  // ---- build CSR (reused by all 3 layers) ----
  pogcn_zero_i4<<<(N_NODES / 4 + 255) / 256, 256, 0, stream>>>((int4*)rowcur, N_NODES / 4);
  pogcn_count<<<(NNZ_TOTAL + 255) / 256, 256, 0, stream>>>(rows, rowcur, NNZ_TOTAL);
  pogcn_scan1<<<NBLK_SCAN1, 256, 0, stream>>>(rowcur, rowptr, bsum, N_NODES);
  pogcn_scan2<<<1, 256, 0, stream>>>(bsum, NBLK_SCAN1);
  pogcn_scan3<<<(N_NODES + 255) / 256, 256, 0, stream>>>(rowptr, rowcur, bsum,
                                                         N_NODES, NNZ_TOTAL);
  const int numTiles = (NNZ_TOTAL + TILE - 1) / TILE;
  pogcn_fill<<<(numTiles + 1) / 2, BLK, 0, stream>>>(rows, cols, vals, rowcur, csr,
                                                     NNZ_TOTAL);

  // ---- embeddings: cur = acc = concat(user, item) ----
  const int n4tot = N_NODES * DIM / 4;
  const int n4u   = NUM_USERS * DIM / 4;
  pogcn_init<<<(n4tot + 255) / 256, 256, 0, stream>>>((const float4*)ue, (const float4*)ie,
                                                      (float4*)A, (float4*)acc, n4u, n4tot);

  // ---- 3 fused propagation layers (no zero pass, no add pass, no f32 atomics) ----
  float* cur = A;
  float* nxt = B;
  for (int l = 0; l < N_LAYERS; ++l) {
    float s      = (l == N_LAYERS - 1) ? 0.25f : 1.0f;
    int   storeY = (l == N_LAYERS - 1) ? 0 : 1;
    pogcn_gather<<<4096, 256, 0, stream>>>(rowptr, csr, cur, nxt, acc, s, storeY);
    float* tmp = cur; cur = nxt; nxt = tmp;
  }
}